// R_GAT_Layer_73297911874085
// MI455X (gfx1250) — compile-verified
//
#include <hip/hip_runtime.h>
#include <hip/hip_bf16.h>

#define N_NODES 50000
#define D_DIM   128
#define E_EDGES 800000
#define NEG_BIG 1e30f
#define ROWS_PER_BLOCK 128   // 8 row-tiles of 16

typedef __attribute__((ext_vector_type(2))) float v2f;
typedef __attribute__((ext_vector_type(8))) float v8f;
typedef __attribute__((ext_vector_type(4))) unsigned int u32x4;
typedef __attribute__((ext_vector_type(4))) int          i32x4;
typedef __attribute__((ext_vector_type(8))) int          i32x8;

#if defined(__gfx1250__) && __has_builtin(__builtin_amdgcn_tensor_load_to_lds)
#define USE_TDM 1
#else
#define USE_TDM 0
#endif

// ---------- helpers ----------
__device__ __forceinline__ unsigned int enc_ordered(float f) {
    unsigned int b = __float_as_uint(f);
    return (b & 0x80000000u) ? ~b : (b | 0x80000000u);
}
__device__ __forceinline__ float dec_ordered(unsigned int u) {
    unsigned int b = (u & 0x80000000u) ? (u ^ 0x80000000u) : ~u;
    return __uint_as_float(b);
}

#if USE_TDM
// Issue a TDM 2D tile load (f32 elements) into LDS. D# layout per CDNA5 ISA §8.
//   tile:   tile_d0 x tile_d1 elements (row-major, dim0 contiguous)
//   tensor: tensor_d0 x tensor_d1, row stride = stride0 elements
__device__ __forceinline__ void tdm_load_2d_f32(unsigned lds_off, const void* gptr,
                                                unsigned tile_d0, unsigned tile_d1,
                                                unsigned tensor_d0, unsigned tensor_d1,
                                                unsigned stride0) {
    unsigned long long ga = (unsigned long long)gptr;
    u32x4 g0;
    g0[0] = 1u;                                        // count=1 (valid user D#), no gather
    g0[1] = lds_off;                                   // lds_addr (bytes)
    g0[2] = (unsigned)(ga & 0xFFFFFFFFu);              // global_addr[31:0]
    g0[3] = (unsigned)((ga >> 32) & 0x01FFFFFFu)       // global_addr[56:32]
          | 0x80000000u;                               // type=2 ("image") in bits 127:126
    i32x8 g1;
    g1[0] = (int)(2u << 16);                           // data_size=2 (4B), wg_mask=0, no pad
    g1[1] = (int)(tensor_d0 << 16);                    // tensor_dim0[15:0] @ bits 63:48
    g1[2] = (int)((tensor_d0 >> 16) | (tensor_d1 << 16));
    g1[3] = (int)((tensor_d1 >> 16) | (tile_d0 << 16));// tile_dim0 @ bits 127:112
    g1[4] = (int)(tile_d1 & 0xFFFFu);                  // tile_dim1; tile_dim2=0
    g1[5] = (int)stride0;                              // tensor_dim0_stride[31:0]
    g1[6] = 0;                                         // stride0 hi16 | stride1 lo16
    g1[7] = 0;
    i32x4 z4 = {0, 0, 0, 0};                           // groups 2/3 unused for 2D
#if __clang_major__ >= 23
    i32x8 z8 = {0, 0, 0, 0, 0, 0, 0, 0};
    __builtin_amdgcn_tensor_load_to_lds(g0, g1, z4, z4, z8, 0);
#else
    __builtin_amdgcn_tensor_load_to_lds(g0, g1, z4, z4, 0);
#endif
}
#endif

// ---------- kernel 0: u = Wr @ wa2  (u[k] = sum_j Wr[k][j] * wa[D+j]) ----------
__global__ void k_compute_u(const float* __restrict__ Wr,
                            const float* __restrict__ wa,
                            float* __restrict__ u) {
    int k = threadIdx.x;
    if (k < D_DIM) {
        float acc = 0.f;
        const float* row = Wr + k * D_DIM;
        const float* wa2 = wa + D_DIM;
        #pragma unroll 8
        for (int j = 0; j < D_DIM; ++j) acc += row[j] * wa2[j];
        u[k] = acc;
    }
}

// ---------- kernel 1: V = x @ Wv + bv   (WMMA f32 16x16x4) ----------
// 256 threads = 8 waves; block covers 128 rows x 128 cols.
// Wave w computes row-tile w across all 8 column tiles (8 x v8f accumulators).
// Wv (64KB) + x tile (64KB) staged in LDS via TDM (wave 0), fallback: vector loads.
__global__ void __launch_bounds__(256)
k_gemm_v_wmma(const float* __restrict__ x,
              const float* __restrict__ Wv,
              const float* __restrict__ bv,
              float* __restrict__ V) {
    __shared__ float lds_wv[D_DIM * D_DIM];           // 64 KB, [k][n]
    __shared__ float lds_x[ROWS_PER_BLOCK * D_DIM];   // 64 KB, [m][k]
    __shared__ float lds_bv[D_DIM];

    const int tid  = threadIdx.x;
    const int r0   = blockIdx.x * ROWS_PER_BLOCK;
    const int rows = min(ROWS_PER_BLOCK, N_NODES - r0);   // 128, or 80 in last block

    if (tid < D_DIM) lds_bv[tid] = bv[tid];

#if USE_TDM
    if (tid < 32) {   // wave 0 issues both DMAs; TENSORcnt tracks them in-order
        tdm_load_2d_f32((unsigned)(unsigned long long)(void*)lds_wv, Wv,
                        D_DIM, D_DIM, D_DIM, D_DIM, D_DIM);
        tdm_load_2d_f32((unsigned)(unsigned long long)(void*)lds_x,
                        x + (size_t)r0 * D_DIM,
                        D_DIM, (unsigned)rows, D_DIM, (unsigned)rows, D_DIM);
        __builtin_amdgcn_s_wait_tensorcnt((short)0);
    }
#else
    {
        const float4* gw = (const float4*)Wv;
        float4*       sw = (float4*)lds_wv;
        #pragma unroll
        for (int i = 0; i < 16; ++i) sw[tid + i * 256] = gw[tid + i * 256];
        const float4* gx = (const float4*)(x + (size_t)r0 * D_DIM);
        float4*       sx = (float4*)lds_x;
        const int nf4 = rows * (D_DIM / 4);
        for (int i = tid; i < nf4; i += 256) sx[i] = gx[i];
    }
#endif
    __syncthreads();

    const int wave      = tid >> 5;
    const int lane      = tid & 31;
    const int tile_rows = rows >> 4;          // 8 (or 5 in last block)
    if (wave < tile_rows) {                   // wave-uniform: EXEC all-ones inside
        const int half = lane >> 4;           // selects K pair {0,1} vs {2,3}
        const int l16  = lane & 15;           // M (A) / N (B) index
        const int koff = half * 2;
        const float* ax = lds_x + (size_t)(wave * 16 + l16) * D_DIM;

        v8f acc[8] = {};
        for (int kk = 0; kk < D_DIM; kk += 4) {
            v2f a;
            a.x = ax[kk + koff];
            a.y = ax[kk + koff + 1];
            const float* b0 = lds_wv + (kk + koff)     * D_DIM + l16;
            const float* b1 = lds_wv + (kk + koff + 1) * D_DIM + l16;
            #pragma unroll
            for (int ct = 0; ct < 8; ++ct) {  // reuse A across 8 column tiles
                v2f b;
                b.x = b0[ct * 16];
                b.y = b1[ct * 16];
                acc[ct] = __builtin_amdgcn_wmma_f32_16x16x4_f32(
                    false, a, false, b, (short)0, acc[ct], false, false);
            }
        }

        // D layout: VGPR j -> lanes 0..15: M=j, lanes 16..31: M=j+8; N = lane&15
        const int rbase = r0 + wave * 16 + half * 8;
        #pragma unroll
        for (int ct = 0; ct < 8; ++ct) {
            const int col  = ct * 16 + l16;
            const float bb = lds_bv[col];
            #pragma unroll
            for (int j = 0; j < 8; ++j) {
                V[(size_t)(rbase + j) * D_DIM + col] = acc[ct][j] + bb;
            }
        }
    }
}

// ---------- kernel 2: per-node GEMVs + init ----------
// one wave per node; 8 waves / block; grid = N/8
__global__ void __launch_bounds__(256)
k_node_init(const float* __restrict__ x,
            const float* __restrict__ wa,   // wa1 = wa[0:D]
            const float* __restrict__ u,
            float* __restrict__ s_tgt,
            float* __restrict__ s_src,
            unsigned int* __restrict__ m_enc,
            float* __restrict__ denom,
            float* __restrict__ out) {
    const int node = blockIdx.x * 8 + (threadIdx.x >> 5);
    const int lane = threadIdx.x & 31;
    if (node >= N_NODES) return;

    const float4 xv = ((const float4*)(x + (size_t)node * D_DIM))[lane];
    const float4 w1 = ((const float4*)wa)[lane];
    const float4 uv = ((const float4*)u)[lane];

    float st = xv.x * w1.x + xv.y * w1.y + xv.z * w1.z + xv.w * w1.w;
    float ss = xv.x * uv.x + xv.y * uv.y + xv.z * uv.z + xv.w * uv.w;
    #pragma unroll
    for (int off = 16; off > 0; off >>= 1) {
        st += __shfl_xor(st, off, 32);
        ss += __shfl_xor(ss, off, 32);
    }
    if (lane == 0) {
        s_tgt[node] = st;
        s_src[node] = ss;
        m_enc[node] = enc_ordered(-NEG_BIG);
        denom[node] = 0.f;
    }
    ((float4*)(out + (size_t)node * D_DIM))[lane] = xv;   // out = x (residual)
}

// ---------- kernel 3: edge scores + segment max ----------
__global__ void __launch_bounds__(256)
k_edge_score(const int* __restrict__ pairs,
             const float* __restrict__ labels,
             const float* __restrict__ s_tgt,
             const float* __restrict__ s_src,
             const float* __restrict__ ba,
             float* __restrict__ e_ws,
             unsigned int* __restrict__ m_enc) {
    const int i = blockIdx.x * blockDim.x + threadIdx.x;
    if (i >= E_EDGES) return;
    const int s = pairs[2 * i];
    const int t = pairs[2 * i + 1];
    float e = s_tgt[t] + s_src[s] + ba[0];
    e = (e > 0.f) ? e : 0.2f * e;          // leaky_relu(0.2)
    e_ws[i] = e;
    if (labels[i] > 0.5f) {
        atomicMax(&m_enc[t], enc_ordered(e));
    }
}

// ---------- kernel 4: z = exp(e-m)*valid ; denom += z ----------
__global__ void __launch_bounds__(256)
k_edge_z(const int* __restrict__ pairs,
         const float* __restrict__ labels,
         const float* __restrict__ e_ws,
         const unsigned int* __restrict__ m_enc,
         float* __restrict__ z_ws,
         float* __restrict__ denom) {
    const int i = blockIdx.x * blockDim.x + threadIdx.x;
    if (i >= E_EDGES) return;
    const int t = pairs[2 * i + 1];
    float z = 0.f;
    if (labels[i] > 0.5f) {
        const float m = dec_ordered(m_enc[t]);   // >= -NEG_BIG by init
        z = __expf(e_ws[i] - m);
        atomicAdd(&denom[t], z);
    }
    z_ws[i] = z;
}

// ---------- kernel 5: out[tgt] += attn * V[src] ----------
// one wave per edge; 8 waves/block; grid = E/8
__global__ void __launch_bounds__(256)
k_edge_agg(const int* __restrict__ pairs,
           const float* __restrict__ z_ws,
           const float* __restrict__ denom,
           const float* __restrict__ V,
           float* __restrict__ out) {
    const int i = blockIdx.x * 8 + (threadIdx.x >> 5);
    const int lane = threadIdx.x & 31;
    if (i >= E_EDGES) return;

    const float z = z_ws[i];
    if (z == 0.f) return;                       // invalid edge
    const int s = pairs[2 * i];
    const int t = pairs[2 * i + 1];
    const float attn = z / fmaxf(denom[t], 1e-30f);

    const float4 v = ((const float4*)(V + (size_t)s * D_DIM))[lane];
    float* o = out + (size_t)t * D_DIM + lane * 4;
    atomicAdd(o + 0, attn * v.x);
    atomicAdd(o + 1, attn * v.y);
    atomicAdd(o + 2, attn * v.z);
    atomicAdd(o + 3, attn * v.w);
}

// ---------- launch ----------
extern "C" void kernel_launch(void* const* d_in, const int* in_sizes, int n_in,
                              void* d_out, int out_size, void* d_ws, size_t ws_size,
                              hipStream_t stream) {
    const float* x      = (const float*)d_in[0];
    const float* labels = (const float*)d_in[1];
    const float* Wv     = (const float*)d_in[2];
    const float* bv     = (const float*)d_in[3];
    const float* Wr     = (const float*)d_in[4];
    const float* wa     = (const float*)d_in[5];
    const float* ba     = (const float*)d_in[6];
    const int*   pairs  = (const int*)d_in[7];
    float* out = (float*)d_out;

    // workspace layout
    float*        V     = (float*)d_ws;                                  // N*D
    unsigned int* m_enc = (unsigned int*)(V + (size_t)N_NODES * D_DIM);  // N
    float*        denom = (float*)(m_enc + N_NODES);                     // N
    float*        s_tgt = denom + N_NODES;                               // N
    float*        s_src = s_tgt + N_NODES;                               // N
    float*        u     = s_src + N_NODES;                               // D
    float*        e_ws  = u + D_DIM;                                     // E
    float*        z_ws  = e_ws + E_EDGES;                                // E

    hipLaunchKernelGGL(k_compute_u, dim3(1), dim3(128), 0, stream, Wr, wa, u);

    const int row_blocks = (N_NODES + ROWS_PER_BLOCK - 1) / ROWS_PER_BLOCK; // 391
    hipLaunchKernelGGL(k_gemm_v_wmma, dim3(row_blocks), dim3(256), 0, stream,
                       x, Wv, bv, V);

    hipLaunchKernelGGL(k_node_init, dim3(N_NODES / 8), dim3(256), 0, stream,
                       x, wa, u, s_tgt, s_src, m_enc, denom, out);

    hipLaunchKernelGGL(k_edge_score, dim3(E_EDGES / 256), dim3(256), 0, stream,
                       pairs, labels, s_tgt, s_src, ba, e_ws, m_enc);

    hipLaunchKernelGGL(k_edge_z, dim3(E_EDGES / 256), dim3(256), 0, stream,
                       pairs, labels, e_ws, m_enc, z_ws, denom);

    hipLaunchKernelGGL(k_edge_agg, dim3(E_EDGES / 8), dim3(256), 0, stream,
                       pairs, z_ws, denom, V, out);
}